// FeatureBank_4982162063364
// MI455X (gfx1250) — compile-verified
//
#include <hip/hip_runtime.h>
#include <stdint.h>

// ---------------------------------------------------------------------------
// FeatureBank memory update for MI455X (gfx1250).
//   out            = copy of memory (500000 x 128 f32)   -> 512 MB of traffic
//   out[y[b], :]   = normalize(0.5*(memory[y[b],:] + x[b,:]))
// Bandwidth-bound: use gfx1250 async global<->LDS streaming for the bulk copy,
// wave32 shuffle reduction for the per-row normalize.
// ---------------------------------------------------------------------------

#define COPY_BLOCKS  2048
#define COPY_THREADS 256
#define SLOTS        8   // float4 slots per thread per iteration (32 KB LDS/block)

__global__ __launch_bounds__(COPY_THREADS)
void bank_copy_async(const float4* __restrict__ src, float4* __restrict__ dst,
                     unsigned long long n4) {
    __shared__ float4 tile[COPY_THREADS * SLOTS];
    const unsigned tid = threadIdx.x;
    // Low 32 bits of the flat shared address == workgroup-relative LDS byte
    // offset (LDS aperture maps offset into addr[31:0]).
    const unsigned ldsBase = (unsigned)(uintptr_t)&tile[0];

    const unsigned long long chunk  = (unsigned long long)(COPY_THREADS * SLOTS);
    const unsigned long long stride = (unsigned long long)gridDim.x * chunk;

    for (unsigned long long base = (unsigned long long)blockIdx.x * chunk;
         base < n4; base += stride) {
        // Phase 1: issue async global->LDS loads (16B per lane per slot).
#pragma unroll
        for (int k = 0; k < SLOTS; ++k) {
            unsigned long long i = base + (unsigned)(k * COPY_THREADS) + tid;
            if (i < n4) {
                const float4* g = src + i;
                unsigned l = ldsBase + (unsigned)((k * COPY_THREADS + tid) * sizeof(float4));
                asm volatile("global_load_async_to_lds_b128 %0, %1, off th:TH_LOAD_NT"
                             :: "v"(l), "v"(g) : "memory");
            }
        }
        asm volatile("s_wait_asynccnt 0x0" ::: "memory");
        // Phase 2: issue async LDS->global stores from the same slots.
#pragma unroll
        for (int k = 0; k < SLOTS; ++k) {
            unsigned long long i = base + (unsigned)(k * COPY_THREADS) + tid;
            if (i < n4) {
                float4* g = dst + i;
                unsigned l = ldsBase + (unsigned)((k * COPY_THREADS + tid) * sizeof(float4));
                asm volatile("global_store_async_from_lds_b128 %0, %1, off th:TH_STORE_NT"
                             :: "v"(g), "v"(l) : "memory");
            }
        }
        // Stores must have drained LDS before next iteration overwrites slots.
        asm volatile("s_wait_asynccnt 0x0" ::: "memory");
    }
}

// One wave32 per row; each lane owns 4 contiguous floats of the 128-wide row.
__global__ __launch_bounds__(256)
void bank_update(const float* __restrict__ x, const long long* __restrict__ y,
                 const float* __restrict__ mem, float* __restrict__ out, int nrows) {
    const int row  = blockIdx.x * 8 + (threadIdx.x >> 5);
    const int lane = threadIdx.x & 31;
    if (row >= nrows) return;

    const long long r = y[row];
    const float4 m  = *(const float4*)(mem + r * 128 + lane * 4);
    const float4 xv = *(const float4*)(x + (long long)row * 128 + lane * 4);

    float4 w;
    w.x = 0.5f * (m.x + xv.x);
    w.y = 0.5f * (m.y + xv.y);
    w.z = 0.5f * (m.z + xv.z);
    w.w = 0.5f * (m.w + xv.w);

    float s = fmaf(w.x, w.x, fmaf(w.y, w.y, fmaf(w.z, w.z, w.w * w.w)));
    // wave32 butterfly reduction (warpSize == 32 on gfx1250)
#pragma unroll
    for (int off = 16; off > 0; off >>= 1)
        s += __shfl_xor(s, off, 32);

    float inv = rsqrtf(s);
    inv = inv * (1.5f - 0.5f * s * inv * inv);   // one Newton step -> ~full f32 precision

    w.x *= inv; w.y *= inv; w.z *= inv; w.w *= inv;
    *(float4*)(out + r * 128 + lane * 4) = w;
}

extern "C" void kernel_launch(void* const* d_in, const int* in_sizes, int n_in,
                              void* d_out, int out_size, void* d_ws, size_t ws_size,
                              hipStream_t stream) {
    const float*     x   = (const float*)d_in[0];      // [4096, 128] f32
    const long long* y   = (const long long*)d_in[1];  // [4096] i64 (unique)
    const float*     mem = (const float*)d_in[2];      // [500000, 128] f32
    float*           out = (float*)d_out;              // [500000, 128] f32

    const int batch = in_sizes[1];                                   // 4096
    const unsigned long long total = (unsigned long long)in_sizes[2]; // 64,000,000
    const unsigned long long n4 = total / 4;                          // float4 count

    // 1) stream memory -> out via async global<->LDS path
    bank_copy_async<<<COPY_BLOCKS, COPY_THREADS, 0, stream>>>(
        (const float4*)mem, (float4*)out, n4);

    // 2) overwrite the 4096 touched rows with the normalized EMA blend
    bank_update<<<(batch + 7) / 8, 256, 0, stream>>>(x, y, mem, out, batch);
}